// Net1_22273700397722
// MI455X (gfx1250) — compile-verified
//
#include <hip/hip_runtime.h>
#include <math.h>

// ---- problem constants (match reference) ----
#define NN     50000
#define EE     800000
#define F_IN0  16
#define EDIM   4
#define EMB    32
#define NH     4
#define HC     128      // EMB*NH
#define DENSE  128

typedef float v2f __attribute__((ext_vector_type(2)));
typedef float v8f __attribute__((ext_vector_type(8)));

// ---- monotone float<->uint mapping for atomic max on signed floats ----
__device__ __forceinline__ unsigned f_enc(float f) {
    unsigned b = __float_as_uint(f);
    return (b & 0x80000000u) ? ~b : (b | 0x80000000u);
}
__device__ __forceinline__ float f_dec(unsigned u) {
    unsigned b = (u & 0x80000000u) ? (u ^ 0x80000000u) : ~u;
    return __uint_as_float(b);
}

// =====================================================================
// Y[nrows x C] = X[nrows x K] @ W[K x C] + bias (optional relu)
// One 16x16 output tile per wave, fp32 WMMA, K stepped by 4.
// nrows % 16 == 0, C % 16 == 0, K % 4 == 0 (all true for this net).
// =====================================================================
__global__ void gemm_bias_wmma(const float* __restrict__ X,
                               const float* __restrict__ W,
                               const float* __restrict__ bias,
                               float* __restrict__ Y,
                               int nrows, int K, int C, int do_relu)
{
    const int wave   = threadIdx.x >> 5;
    const int lane   = threadIdx.x & 31;
    const int ctiles = C >> 4;
    const int tiles  = (nrows >> 4) * ctiles;
    const int t      = blockIdx.x * (blockDim.x >> 5) + wave;
    if (t >= tiles) return;                   // uniform per wave -> EXEC stays full
    const int row0 = (t / ctiles) << 4;
    const int col0 = (t % ctiles) << 4;
    const int half = lane >> 4;               // 0: K-pair {0,1}, 1: K-pair {2,3}
    const int r    = lane & 15;

    v8f acc = {};
    const float* xrow = X + (size_t)(row0 + r) * K;
    for (int k0 = 0; k0 < K; k0 += 4) {
        v2f a, b;
        a.x = xrow[k0 + 2 * half + 0];
        a.y = xrow[k0 + 2 * half + 1];
        const float* wp = W + (size_t)(k0 + 2 * half) * C + col0 + r;
        b.x = wp[0];
        b.y = wp[C];
        acc = __builtin_amdgcn_wmma_f32_16x16x4_f32(false, a, false, b,
                                                    (short)0, acc, false, false);
    }
    const float bv = bias[col0 + r];
#pragma unroll
    for (int i = 0; i < 8; ++i) {             // D: lane<16 -> M=i, lane>=16 -> M=i+8
        int m = i + 8 * half;
        float val = acc[i] + bv;
        if (do_relu) val = fmaxf(val, 0.0f);
        Y[(size_t)(row0 + m) * C + col0 + r] = val;
    }
}

// =====================================================================
// init per-(node,head) softmax state: m = enc(-inf), z = 0
// =====================================================================
__global__ void init_mz(unsigned* __restrict__ m, float* __restrict__ z, int n)
{
    int i = blockIdx.x * blockDim.x + threadIdx.x;
    if (i < n) { m[i] = 0x007FFFFFu; z[i] = 0.0f; }   // enc(-inf)
}

// =====================================================================
// Pass 1: per (edge, head) logits + segment max (edge embedding inline)
// =====================================================================
__global__ void edge_logits(const float* __restrict__ q,
                            const float* __restrict__ k,
                            const float* __restrict__ ea,
                            const int*   __restrict__ eidx,
                            const float* __restrict__ We,
                            float*       __restrict__ logits,
                            unsigned*    __restrict__ mbuf)
{
    int gid = blockIdx.x * blockDim.x + threadIdx.x;
    if (gid >= EE * NH) return;
    int e = gid >> 2, h = gid & 3;
    int src = eidx[e];
    int dst = eidx[EE + e];
    float a0 = ea[e * 4 + 0], a1 = ea[e * 4 + 1];
    float a2 = ea[e * 4 + 2], a3 = ea[e * 4 + 3];
    const float* qp = q + (size_t)dst * HC + h * EMB;
    const float* kp = k + (size_t)src * HC + h * EMB;
    const float* wp = We + h * EMB;
    float dot = 0.0f;
#pragma unroll 8
    for (int j = 0; j < EMB; ++j) {
        float ej = a0 * wp[0 * HC + j] + a1 * wp[1 * HC + j]
                 + a2 * wp[2 * HC + j] + a3 * wp[3 * HC + j];
        dot += qp[j] * (kp[j] + ej);
    }
    float logit = dot * 0.17677669529663687f;   // 1/sqrt(32)
    logits[gid] = logit;
    atomicMax(mbuf + (size_t)dst * NH + h, f_enc(logit));
}

// =====================================================================
// Pass 2: p = exp(l - m[dst]); z[dst] += p
// =====================================================================
__global__ void edge_exp(float* __restrict__ logits,
                         const unsigned* __restrict__ mbuf,
                         float* __restrict__ z,
                         const int* __restrict__ eidx)
{
    int gid = blockIdx.x * blockDim.x + threadIdx.x;
    if (gid >= EE * NH) return;
    int e = gid >> 2, h = gid & 3;
    int dst = eidx[EE + e];
    float mf = f_dec(mbuf[(size_t)dst * NH + h]);
    float p = __expf(logits[gid] - mf);
    logits[gid] = p;
    atomicAdd(z + (size_t)dst * NH + h, p);
}

// =====================================================================
// Pass 3: acc[dst] += (v[src] + e) * p/(z[dst]+1e-16)
// =====================================================================
__global__ void edge_scatter(const float* __restrict__ logits,
                             const float* __restrict__ z,
                             const float* __restrict__ v,
                             const float* __restrict__ ea,
                             const int*   __restrict__ eidx,
                             const float* __restrict__ We,
                             float*       __restrict__ acc)
{
    int gid = blockIdx.x * blockDim.x + threadIdx.x;
    if (gid >= EE * NH) return;
    int e = gid >> 2, h = gid & 3;
    int src = eidx[e];
    int dst = eidx[EE + e];
    float alpha = logits[gid] / (z[(size_t)dst * NH + h] + 1e-16f);
    float a0 = ea[e * 4 + 0], a1 = ea[e * 4 + 1];
    float a2 = ea[e * 4 + 2], a3 = ea[e * 4 + 3];
    const float* wp = We + h * EMB;
    const float* vp = v + (size_t)src * HC + h * EMB;
    float* ap = acc + (size_t)dst * HC + h * EMB;
#pragma unroll 8
    for (int j = 0; j < EMB; ++j) {
        float ej = a0 * wp[0 * HC + j] + a1 * wp[1 * HC + j]
                 + a2 * wp[2 * HC + j] + a3 * wp[3 * HC + j];
        atomicAdd(ap + j, (vp[j] + ej) * alpha);
    }
}

// =====================================================================
// Fused dense head: 32 ->(relu) 128 ->(relu) 64 ->(sigmoid) 1
// One wave per 16 nodes; WMMA stages with LDS intermediates.
// =====================================================================
#define HEAD_WPB 4
__global__ void head_kernel(const float* __restrict__ h,
                            const float* __restrict__ W1, const float* __restrict__ b1,
                            const float* __restrict__ W2, const float* __restrict__ b2,
                            const float* __restrict__ W3, const float* __restrict__ b3,
                            float* __restrict__ out, int nrows)
{
    __shared__ float Y1[HEAD_WPB][16 * 128];
    __shared__ float Y2[HEAD_WPB][16 * 64];
    const int wave = threadIdx.x >> 5;
    const int lane = threadIdx.x & 31;
    const int half = lane >> 4;
    const int r    = lane & 15;
    const int node0 = (blockIdx.x * HEAD_WPB + wave) << 4;
    if (node0 >= nrows) return;              // uniform per wave
    float* y1 = Y1[wave];
    float* y2 = Y2[wave];

    // stage 1: h[16x32] @ W1[32x128] + b1, relu -> LDS
    for (int ct = 0; ct < 8; ++ct) {
        v8f acc = {};
        for (int k0 = 0; k0 < 32; k0 += 4) {
            v2f a, b;
            const float* hp = h + (size_t)(node0 + r) * 32 + k0 + 2 * half;
            a.x = hp[0]; a.y = hp[1];
            const float* wp = W1 + (size_t)(k0 + 2 * half) * 128 + ct * 16 + r;
            b.x = wp[0]; b.y = wp[128];
            acc = __builtin_amdgcn_wmma_f32_16x16x4_f32(false, a, false, b,
                                                        (short)0, acc, false, false);
        }
        float bv = b1[ct * 16 + r];
#pragma unroll
        for (int i = 0; i < 8; ++i) {
            int m = i + 8 * half;
            y1[m * 128 + ct * 16 + r] = fmaxf(acc[i] + bv, 0.0f);
        }
    }
    asm volatile("s_wait_dscnt 0" ::: "memory");   // wave-local LDS RAW

    // stage 2: Y1[16x128] @ W2[128x64] + b2, relu -> LDS
    for (int ct = 0; ct < 4; ++ct) {
        v8f acc = {};
        for (int k0 = 0; k0 < 128; k0 += 4) {
            v2f a, b;
            a.x = y1[r * 128 + k0 + 2 * half];
            a.y = y1[r * 128 + k0 + 2 * half + 1];
            const float* wp = W2 + (size_t)(k0 + 2 * half) * 64 + ct * 16 + r;
            b.x = wp[0]; b.y = wp[64];
            acc = __builtin_amdgcn_wmma_f32_16x16x4_f32(false, a, false, b,
                                                        (short)0, acc, false, false);
        }
        float bv = b2[ct * 16 + r];
#pragma unroll
        for (int i = 0; i < 8; ++i) {
            int m = i + 8 * half;
            y2[m * 64 + ct * 16 + r] = fmaxf(acc[i] + bv, 0.0f);
        }
    }
    asm volatile("s_wait_dscnt 0" ::: "memory");

    // stage 3: Y2[16x64] @ W3[64x1] + b3, sigmoid
    if (half == 0) {
        float s = b3[0];
#pragma unroll 8
        for (int j = 0; j < 64; ++j) s += y2[r * 64 + j] * W3[j];
        out[node0 + r] = 1.0f / (1.0f + __expf(-s));
    }
}

// =====================================================================
// host orchestration
// =====================================================================
extern "C" void kernel_launch(void* const* d_in, const int* in_sizes, int n_in,
                              void* d_out, int out_size, void* d_ws, size_t ws_size,
                              hipStream_t stream)
{
    (void)in_sizes; (void)n_in; (void)out_size; (void)ws_size;

    const float* x     = (const float*)d_in[0];
    const int*   eidx  = (const int*)  d_in[1];
    const float* eattr = (const float*)d_in[2];
    const float* Wq0 = (const float*)d_in[3];  const float* bq0 = (const float*)d_in[4];
    const float* Wk0 = (const float*)d_in[5];  const float* bk0 = (const float*)d_in[6];
    const float* Wv0 = (const float*)d_in[7];  const float* bv0 = (const float*)d_in[8];
    const float* We0 = (const float*)d_in[9];
    const float* Ws0 = (const float*)d_in[10]; const float* bs0 = (const float*)d_in[11];
    const float* Wt0 = (const float*)d_in[12]; const float* bt0 = (const float*)d_in[13];
    const float* WqL = (const float*)d_in[14]; const float* bqL = (const float*)d_in[15];
    const float* WkL = (const float*)d_in[16]; const float* bkL = (const float*)d_in[17];
    const float* WvL = (const float*)d_in[18]; const float* bvL = (const float*)d_in[19];
    const float* WeL = (const float*)d_in[20];
    const float* WsL = (const float*)d_in[21]; const float* bsL = (const float*)d_in[22];
    const float* WtL = (const float*)d_in[23]; const float* btL = (const float*)d_in[24];
    const float* W1  = (const float*)d_in[25]; const float* b1  = (const float*)d_in[26];
    const float* W2  = (const float*)d_in[27]; const float* b2  = (const float*)d_in[28];
    const float* W3  = (const float*)d_in[29]; const float* b3  = (const float*)d_in[30];

    // workspace layout (floats): ~123 MB total
    float* q      = (float*)d_ws;
    float* kbuf   = q    + (size_t)NN * HC;
    float* vbuf   = kbuf + (size_t)NN * HC;
    float* accb   = vbuf + (size_t)NN * HC;   // skip + aggregated messages
    float* hbuf   = accb + (size_t)NN * HC;   // per-layer node features (N x 32)
    float* logits = hbuf + (size_t)NN * EMB;  // E x H (reused as p)
    unsigned* mbuf = (unsigned*)(logits + (size_t)EE * NH);
    float* zbuf   = (float*)(mbuf + (size_t)NN * NH);

    const int WV = 8;                          // waves per gemm block
    dim3 blk(256);
    auto cdiv = [](int a, int b) { return (a + b - 1) / b; };

    for (int layer = 0; layer < 3; ++layer) {
        const float *Wq_, *bq_, *Wk_, *bk_, *Wv_, *bv_, *We_, *Ws_, *bs_, *Wt_, *bt_;
        const float* X; int F;
        if (layer == 0) {
            Wq_ = Wq0; bq_ = bq0; Wk_ = Wk0; bk_ = bk0; Wv_ = Wv0; bv_ = bv0;
            We_ = We0; Ws_ = Ws0; bs_ = bs0; Wt_ = Wt0; bt_ = bt0;
            X = x; F = F_IN0;
        } else {
            int i = layer - 1;
            Wq_ = WqL + (size_t)i * EMB * HC;  bq_ = bqL + (size_t)i * HC;
            Wk_ = WkL + (size_t)i * EMB * HC;  bk_ = bkL + (size_t)i * HC;
            Wv_ = WvL + (size_t)i * EMB * HC;  bv_ = bvL + (size_t)i * HC;
            We_ = WeL + (size_t)i * EDIM * HC;
            Ws_ = WsL + (size_t)i * EMB * HC;  bs_ = bsL + (size_t)i * HC;
            Wt_ = WtL + (size_t)i * HC * EMB;  bt_ = btL + (size_t)i * EMB;
            X = hbuf; F = EMB;
        }
        const int tQ = (NN / 16) * (HC / 16);  // 25000 tiles
        gemm_bias_wmma<<<cdiv(tQ, WV), blk, 0, stream>>>(X, Wq_, bq_, q,    NN, F, HC, 0);
        gemm_bias_wmma<<<cdiv(tQ, WV), blk, 0, stream>>>(X, Wk_, bk_, kbuf, NN, F, HC, 0);
        gemm_bias_wmma<<<cdiv(tQ, WV), blk, 0, stream>>>(X, Wv_, bv_, vbuf, NN, F, HC, 0);
        gemm_bias_wmma<<<cdiv(tQ, WV), blk, 0, stream>>>(X, Ws_, bs_, accb, NN, F, HC, 0);

        init_mz<<<cdiv(NN * NH, 256), 256, 0, stream>>>(mbuf, zbuf, NN * NH);

        const int eg = cdiv(EE * NH, 256);
        edge_logits <<<eg, 256, 0, stream>>>(q, kbuf, eattr, eidx, We_, logits, mbuf);
        edge_exp    <<<eg, 256, 0, stream>>>(logits, mbuf, zbuf, eidx);
        edge_scatter<<<eg, 256, 0, stream>>>(logits, zbuf, vbuf, eattr, eidx, We_, accb);

        const int tT = (NN / 16) * (EMB / 16);
        gemm_bias_wmma<<<cdiv(tT, WV), blk, 0, stream>>>(accb, Wt_, bt_, hbuf, NN, HC, EMB, 1);
    }

    head_kernel<<<cdiv(NN / 16, HEAD_WPB), HEAD_WPB * 32, 0, stream>>>(
        hbuf, W1, b1, W2, b2, W3, b3, (float*)d_out, NN);
}